// SingleAttentionHead_36086315221265
// MI455X (gfx1250) — compile-verified
//
#include <hip/hip_runtime.h>

typedef float v2f __attribute__((ext_vector_type(2)));
typedef float v4f __attribute__((ext_vector_type(4)));
typedef float v8f __attribute__((ext_vector_type(8)));

#define WMMA_F32(a, b, c) \
    __builtin_amdgcn_wmma_f32_16x16x4_f32(false, (a), false, (b), (short)0, (c), false, false)

constexpr int Bn = 8, Tn = 2048, Cn = 1024, Hn = 64;

// workspace layout (in floats)
constexpr size_t OFF_KP = 0;                              // [B][T][H] row-major
constexpr size_t OFF_QP = OFF_KP + (size_t)Bn * Tn * Hn;  // [B][T][H], pre-scaled by 1/8
constexpr size_t OFF_VT = OFF_QP + (size_t)Bn * Tn * Hn;  // [B][H][T] (transposed)
constexpr size_t OFF_WT = OFF_VT + (size_t)Bn * Tn * Hn;  // 3 x [H][C] transposed weights

// ---------------------------------------------------------------------------
// Kernel 1: transpose weights to [H][C]; fold attention scale (1/8) into Wq.
// ---------------------------------------------------------------------------
__global__ void wtrans_kernel(const float* __restrict__ Wk,
                              const float* __restrict__ Wq,
                              const float* __restrict__ Wv,
                              float* __restrict__ wt) {
    int idx = blockIdx.x * 256 + threadIdx.x;   // 0 .. 3*65536-1 (exact)
    int m = idx >> 16;                          // which matrix
    int e = idx & 65535;                        // element within [H][C]
    int h = e >> 10;
    int c = e & 1023;
    const float* W = (m == 0) ? Wk : ((m == 1) ? Wq : Wv);
    float val = W[c * Hn + h];
    if (m == 1) val *= 0.125f;                  // HEAD_SIZE^-0.5
    wt[(size_t)m * 65536 + e] = val;
}

// ---------------------------------------------------------------------------
// Kernel 2: projections. One wave computes a 16x64 output tile of k, q or v.
// A-frag: x rows (contiguous float2). B-frag: Wt rows (contiguous float2).
// f32 16x4 A layout: lanes 0-15 hold K=k0,k0+1 of row M=lane; lanes 16-31
// hold K=k0+2,k0+3 of row M=lane-16 (VGPR0=K even slot, VGPR1=odd).
// ---------------------------------------------------------------------------
__global__ void proj_kernel(const float* __restrict__ x,
                            const float* __restrict__ wt,
                            float* __restrict__ ws) {
    int wave  = blockIdx.x * (blockDim.x >> 5) + (threadIdx.x >> 5);
    int lane  = threadIdx.x & 31;
    int which = wave % 3;            // 0=k, 1=q, 2=v
    int rt    = wave / 3;            // row tile 0..1023 over B*T
    int l15   = lane & 15;
    int hi    = lane >> 4;

    const float* xa  = x + (size_t)(rt * 16 + l15) * Cn + 2 * hi;
    const float* wbb = wt + (size_t)which * Hn * Cn + 2 * hi;
    const float* wb0 = wbb + (size_t)(0 * 16 + l15) * Cn;
    const float* wb1 = wbb + (size_t)(1 * 16 + l15) * Cn;
    const float* wb2 = wbb + (size_t)(2 * 16 + l15) * Cn;
    const float* wb3 = wbb + (size_t)(3 * 16 + l15) * Cn;

    v8f a0 = {}, a1 = {}, a2 = {}, a3 = {};
#pragma unroll 4
    for (int k = 0; k < Cn; k += 4) {
        v2f a = *(const v2f*)(xa + k);
        a0 = WMMA_F32(a, *(const v2f*)(wb0 + k), a0);
        a1 = WMMA_F32(a, *(const v2f*)(wb1 + k), a1);
        a2 = WMMA_F32(a, *(const v2f*)(wb2 + k), a2);
        a3 = WMMA_F32(a, *(const v2f*)(wb3 + k), a3);
    }

    // C/D layout: lane holds column N=l15; VGPR r holds row M=r+8*hi.
    if (which < 2) {
        float* p = ws + (which == 0 ? OFF_KP : OFF_QP);
        int rowbase = rt * 16 + 8 * hi;
#pragma unroll
        for (int r = 0; r < 8; r++) {
            p[(size_t)(rowbase + r) * Hn + 0 * 16 + l15] = a0[r];
            p[(size_t)(rowbase + r) * Hn + 1 * 16 + l15] = a1[r];
            p[(size_t)(rowbase + r) * Hn + 2 * 16 + l15] = a2[r];
            p[(size_t)(rowbase + r) * Hn + 3 * 16 + l15] = a3[r];
        }
    } else {
        // store v transposed: vT[b][h][t]; per lane the 8 rows are consecutive t
        int b  = rt >> 7;
        int t0 = (rt & 127) * 16 + 8 * hi;
        float* vtb = ws + OFF_VT + (size_t)b * Hn * Tn + t0;
        v8f accs0 = a0, accs1 = a1, accs2 = a2, accs3 = a3;
        {
            float* d = vtb + (size_t)(0 * 16 + l15) * Tn;
            *(v4f*)(d)     = (v4f){accs0[0], accs0[1], accs0[2], accs0[3]};
            *(v4f*)(d + 4) = (v4f){accs0[4], accs0[5], accs0[6], accs0[7]};
        }
        {
            float* d = vtb + (size_t)(1 * 16 + l15) * Tn;
            *(v4f*)(d)     = (v4f){accs1[0], accs1[1], accs1[2], accs1[3]};
            *(v4f*)(d + 4) = (v4f){accs1[4], accs1[5], accs1[6], accs1[7]};
        }
        {
            float* d = vtb + (size_t)(2 * 16 + l15) * Tn;
            *(v4f*)(d)     = (v4f){accs2[0], accs2[1], accs2[2], accs2[3]};
            *(v4f*)(d + 4) = (v4f){accs2[4], accs2[5], accs2[6], accs2[7]};
        }
        {
            float* d = vtb + (size_t)(3 * 16 + l15) * Tn;
            *(v4f*)(d)     = (v4f){accs3[0], accs3[1], accs3[2], accs3[3]};
            *(v4f*)(d + 4) = (v4f){accs3[4], accs3[5], accs3[6], accs3[7]};
        }
    }
}

// ---------------------------------------------------------------------------
// Kernel 3: flash attention. One wave per (batch, 16-row tile i).
// Reference: wei[t][s] = k[t]·q[s] (roles swapped). We compute the tile
// TRANSPOSED: St[s][t] = q_j-tile x kp_i-tile^T, so softmax over s is an
// in-lane reduction over the 8 C-VGPRs + one xor-16 shuffle. Then
// O^T += V^T @ P^T keeps everything in C-layout; final 1/l is lane-uniform.
// ---------------------------------------------------------------------------
__global__ void attn_kernel(const float* __restrict__ ws, float* __restrict__ out) {
    int wave = blockIdx.x * (blockDim.x >> 5) + (threadIdx.x >> 5);
    int lane = threadIdx.x & 31;
    int b = wave >> 7;        // batch
    int i = wave & 127;       // row tile (t-dimension)
    int l15 = lane & 15;
    int hi  = lane >> 4;

    const float* kp = ws + OFF_KP + (size_t)(b * Tn + i * 16 + l15) * Hn + 2 * hi;
    const float* qp = ws + OFF_QP + (size_t)b * Tn * Hn;
    const float* vt = ws + OFF_VT + (size_t)b * Hn * Tn;

    // B-frags of kp^T for the i-tile are loop-invariant: preload 16 float2.
    v2f kb[16];
#pragma unroll
    for (int kk = 0; kk < 16; kk++) kb[kk] = *(const v2f*)(kp + 4 * kk);

    v8f oacc[4] = {};
    float m = -__builtin_inff();
    float l = 0.0f;

    for (int j = 0; j <= i; j++) {
        // St = q_jtile(16x64) @ kp_itile^T(64x16); D: M=s_local, N=t_local
        const float* qa = qp + (size_t)(j * 16 + l15) * Hn + 2 * hi;
        v8f s = {};
#pragma unroll
        for (int kk = 0; kk < 16; kk++) {
            v2f a = *(const v2f*)(qa + 4 * kk);
            s = WMMA_F32(a, kb[kk], s);
        }

        // causal mask on the diagonal tile: valid iff s_local <= t_local
        if (j == i) {
#pragma unroll
            for (int r = 0; r < 8; r++) {
                int sl = r + 8 * hi;
                s[r] = (sl > l15) ? -__builtin_inff() : s[r];
            }
        }

        // online softmax (per column t; lane and lane^16 share the same t)
        float tm = s[0];
#pragma unroll
        for (int r = 1; r < 8; r++) tm = fmaxf(tm, s[r]);
        tm = fmaxf(tm, __shfl_xor(tm, 16, 32));
        float mn = fmaxf(m, tm);
        float sc = __expf(m - mn);
        m = mn;

        float p[8];
        float tl = 0.0f;
#pragma unroll
        for (int r = 0; r < 8; r++) {
            p[r] = __expf(s[r] - mn);
            tl += p[r];
        }
        tl += __shfl_xor(tl, 16, 32);
        l = l * sc + tl;
#pragma unroll
        for (int n = 0; n < 4; n++) oacc[n] = oacc[n] * sc;

        // Re-fragment P^T (C-layout) into B-layout for the next WMMA:
        // B kstep kk needs K=4kk..4kk+1 (lanes<16) / 4kk+2..4kk+3 (lanes>=16).
        float q0 = __shfl_xor(p[0], 16, 32), q1 = __shfl_xor(p[1], 16, 32);
        float q2 = __shfl_xor(p[2], 16, 32), q3 = __shfl_xor(p[3], 16, 32);
        float q4 = __shfl_xor(p[4], 16, 32), q5 = __shfl_xor(p[5], 16, 32);
        float q6 = __shfl_xor(p[6], 16, 32), q7 = __shfl_xor(p[7], 16, 32);
        bool ishi = (hi != 0);
        v2f bb[4];
        bb[0] = ishi ? (v2f){q2, q3} : (v2f){p[0], p[1]};
        bb[1] = ishi ? (v2f){q6, q7} : (v2f){p[4], p[5]};
        bb[2] = ishi ? (v2f){p[2], p[3]} : (v2f){q0, q1};
        bb[3] = ishi ? (v2f){p[6], p[7]} : (v2f){q4, q5};

        // O^T += V^T(64x16) @ P^T(16x16): A-frags from vT are contiguous.
        const float* va = vt + (size_t)l15 * Tn + j * 16 + 2 * hi;
#pragma unroll
        for (int n = 0; n < 4; n++) {
            const float* van = va + (size_t)n * 16 * Tn;
#pragma unroll
            for (int kk = 0; kk < 4; kk++) {
                v2f a = *(const v2f*)(van + 4 * kk);
                oacc[n] = WMMA_F32(a, bb[kk], oacc[n]);
            }
        }
    }

    // O^T C-layout: lane holds column t=l15; VGPR r holds h = n*16 + r + 8*hi.
    float invl = 1.0f / l;
    float* dst = out + (size_t)(b * Tn + i * 16 + l15) * Hn + 8 * hi;
#pragma unroll
    for (int n = 0; n < 4; n++) {
        v4f lo = {oacc[n][0] * invl, oacc[n][1] * invl, oacc[n][2] * invl, oacc[n][3] * invl};
        v4f hi4 = {oacc[n][4] * invl, oacc[n][5] * invl, oacc[n][6] * invl, oacc[n][7] * invl};
        *(v4f*)(dst + n * 16)     = lo;
        *(v4f*)(dst + n * 16 + 4) = hi4;
    }
}

// ---------------------------------------------------------------------------
extern "C" void kernel_launch(void* const* d_in, const int* in_sizes, int n_in,
                              void* d_out, int out_size, void* d_ws, size_t ws_size,
                              hipStream_t stream) {
    const float* x  = (const float*)d_in[0];
    const float* Wk = (const float*)d_in[1];
    const float* Wq = (const float*)d_in[2];
    const float* Wv = (const float*)d_in[3];
    float* ws  = (float*)d_ws;
    float* out = (float*)d_out;

    // 1) transpose weights (3 * 64 * 1024 elements, 256 threads/block)
    wtrans_kernel<<<768, 256, 0, stream>>>(Wk, Wq, Wv, ws + OFF_WT);
    // 2) projections: 1024 row-tiles * 3 matrices = 3072 waves, 8 waves/block
    proj_kernel<<<384, 256, 0, stream>>>(x, ws + OFF_WT, ws);
    // 3) flash attention: 8 batches * 128 row-tiles = 1024 waves, 8 waves/block
    attn_kernel<<<128, 256, 0, stream>>>(ws, out);
}